// LearnableSparseAdjacency_6098853560443
// MI455X (gfx1250) — compile-verified
//
#include <hip/hip_runtime.h>
#include <hip/hip_bf16.h>
#include <stdint.h>

#define NROW 8192
#define KNBR 64
#define BLOCK 256
#define CHUNKS (NROW / (BLOCK * 4))   // 8 float4 chunks per thread

// Pointer types matching the gfx1250 async-store builtin signature:
// b128 variants take pointers to 4 x i32 vectors in AS1 (global) / AS3 (LDS).
typedef int v4i __attribute__((ext_vector_type(4)));
typedef __attribute__((address_space(1))) v4i* global_v4i_ptr;
typedef __attribute__((address_space(3))) v4i* lds_v4i_ptr;

// gfx12-family CPol: TH in bits [2:0]; TH=1 -> non-temporal (streaming store).
#define CPOL_STORE_NT 1

// Build one dense row per workgroup in LDS, then stream it out.
// Phase 0: issue (idx,val) loads early so HBM latency overlaps the LDS zero fill
// Phase 1: zero 32KiB LDS row (conflict-free lane-contiguous ds_store_b128)
// Phase 2: scatter the 64 (idx, val) pairs into the LDS row
// Phase 3: async LDS->global B128 NT stores (CDNA5 path), 512B/wave/instr
__global__ __launch_bounds__(BLOCK) void sparse_adj_scatter_kernel(
    const float* __restrict__ vals,      // [NROW, KNBR] fp32
    const int*   __restrict__ nbr_idx,   // [NROW, KNBR] int32, unique per row
    float*       __restrict__ out)       // [NROW, NROW] fp32
{
    __shared__ float row[NROW];          // 32 KiB

    const int tid = threadIdx.x;
    const int r   = blockIdx.x;

    // ---- Phase 0: start the scatter-pair loads (overlap with zero fill) --
    int   j = 0;
    float v = 0.f;
    if (tid < KNBR) {
        const size_t base = (size_t)r * KNBR + tid;
        j = nbr_idx[base];
        v = vals[base];
    }

    // ---- Phase 1: zero the LDS row -------------------------------------
    float4* row4 = (float4*)row;
#pragma unroll
    for (int c = 0; c < CHUNKS; ++c) {
        row4[c * BLOCK + tid] = make_float4(0.f, 0.f, 0.f, 0.f);
    }
    __syncthreads();

    // ---- Phase 2: scatter K values into the row ------------------------
    if (tid < KNBR) {
        row[j] = v;                       // indices unique per row -> no conflict
    }
    __syncthreads();

    // ---- Phase 3: stream LDS row to global -----------------------------
    float* orow = out + (size_t)r * NROW;

#if __has_builtin(__builtin_amdgcn_global_store_async_from_lds_b128)
#pragma unroll
    for (int c = 0; c < CHUNKS; ++c) {
        const int e = (c * BLOCK + tid) * 4;   // float index, 16B aligned
        __builtin_amdgcn_global_store_async_from_lds_b128(
            (global_v4i_ptr)(orow + e),
            (lds_v4i_ptr)(row + e),
            /*offset=*/0, /*cpol=*/CPOL_STORE_NT);
    }
#if __has_builtin(__builtin_amdgcn_s_wait_asynccnt)
    __builtin_amdgcn_s_wait_asynccnt(0);
#else
    asm volatile("s_wait_asynccnt 0" ::: "memory");
#endif
#else
    // Fallback: plain wide copy LDS -> global (still full-rate b128 stores).
#pragma unroll
    for (int c = 0; c < CHUNKS; ++c) {
        const int e4 = c * BLOCK + tid;
        ((float4*)orow)[e4] = row4[e4];
    }
#endif
}

extern "C" void kernel_launch(void* const* d_in, const int* in_sizes, int n_in,
                              void* d_out, int out_size, void* d_ws, size_t ws_size,
                              hipStream_t stream) {
    // setup_inputs() order: adjacency_values (fp32 [N,K]), neighbor_indices (i32 [N,K])
    const float* vals = (const float*)d_in[0];
    const int*   idx  = (const int*)d_in[1];
    float*       out  = (float*)d_out;   // [N, N] fp32

    (void)in_sizes; (void)n_in; (void)out_size; (void)d_ws; (void)ws_size;

    sparse_adj_scatter_kernel<<<dim3(NROW), dim3(BLOCK), 0, stream>>>(vals, idx, out);
}